// UnimodalNormal_12678743458349
// MI455X (gfx1250) — compile-verified
//
#include <hip/hip_runtime.h>
#include <math.h>

// CDNA5 / gfx1250 kernel for UnimodalNormal head.
// B=262144 rows, D=512, C=100.
//
// Wave-level plan (wave32):
//   - Each wave owns 16 rows. Accumulate A(16x512) x B(512x2) with
//     V_WMMA_F32_16X16X4_F32 (fp32, K=4), 128 WMMAs per tile.
//     B-matrix column 0 = w_mu, column 1 = w_sig, columns 2..15 = 0,
//     built branchlessly from an LDS-packed weight table (lane masks + FMA).
//   - Row normalization telescopes: sum(probs) = cdf[C] - cdf[0].
//
// Memory-bound: 512MB x stream + 105MB out => ~26.5us floor @ 23.3 TB/s.
// x / out use non-temporal hints to avoid churning the 192MB L2.

typedef __attribute__((ext_vector_type(2))) float v2f;
typedef __attribute__((ext_vector_type(4))) float v4f;
typedef __attribute__((ext_vector_type(8))) float v8f;

#define DHID   512
#define NCAT   100
#define ROWS_PER_BLOCK 128   // 8 waves * 16 rows
#define BLOCK  256

__global__ __launch_bounds__(BLOCK) void unimodal_normal_wmma_kernel(
    const float* __restrict__ x,
    const float* __restrict__ w_mu,
    const float* __restrict__ b_mu,
    const float* __restrict__ w_sig,
    const float* __restrict__ b_sig,
    float* __restrict__ out)
{
    // Packed weights: wpack[2k]=w_mu[k], wpack[2k+1]=w_sig[k] -> 4KB
    __shared__ __align__(16) float wpack[2 * DHID];
    // Per-wave pre-activations: [wave][row*2 + {mu,sig}]
    __shared__ float pre[8][32];
    // Per-wave CDF staging (101 used, padded)
    __shared__ float cdfbuf[8][112];

    const int tid  = threadIdx.x;
    const int lane = tid & 31;
    const int wave = tid >> 5;
    const int half = lane >> 4;   // 0: K=k..k+1, 1: K=k+2..k+3 (A layout)
    const int col  = lane & 15;   // M row index for A; N column index for B

    // Branchless B-fragment lane masks: column 0 = w_mu, column 1 = w_sig.
    const float m0 = (col == 0) ? 1.0f : 0.0f;
    const float m1 = (col == 1) ? 1.0f : 0.0f;

    // Cooperative load of packed weights into LDS.
    for (int t = tid; t < DHID; t += BLOCK) {
        wpack[2 * t + 0] = w_mu[t];
        wpack[2 * t + 1] = w_sig[t];
    }
    __syncthreads();

    const int row0 = blockIdx.x * ROWS_PER_BLOCK + wave * 16;

    // Per-lane base pointer into x for the A-fragment layout:
    // lane (half,col) supplies A[M=col][K = k + 2*half .. +1].
    const float* xrow = x + (size_t)(row0 + col) * DHID + (half << 1);

    v8f acc = {};   // D[16x16]; only columns 0 (mu) and 1 (sig) are meaningful

    for (int k0 = 0; k0 < DHID; k0 += 64) {
        // Speculative stream prefetch ~1KB ahead (global_prefetch_b8).
        __builtin_prefetch(xrow + k0 + 256, 0, 1);
#pragma unroll
        for (int kk = 0; kk < 64; kk += 4) {
            const int k = k0 + kk;
            // A fragment: one non-temporal b64 load per lane.
            v2f A = __builtin_nontemporal_load((const v2f*)(xrow + k));

            // B fragment: VGPR0 = B[2*half][col], VGPR1 = B[2*half+1][col].
            // wpack index 2*(k + 2*half) is multiple of 4 -> 16B-aligned b128.
            v4f wq = *(const v4f*)(&wpack[2 * (k + (half << 1))]);
            v2f Bf;
            Bf.x = m0 * wq.x + m1 * wq.y;
            Bf.y = m0 * wq.z + m1 * wq.w;

            acc = __builtin_amdgcn_wmma_f32_16x16x4_f32(
                /*neg_a=*/false, A, /*neg_b=*/false, Bf,
                /*c_mod=*/(short)0, acc, /*reuse_a=*/false, /*reuse_b=*/false);
        }
    }

    // Extract columns 0/1 of D to LDS.
    // C/D layout: VGPR v, lanes 0-15 -> M=v, N=lane; lanes 16-31 -> M=v+8.
    if (col < 2) {
#pragma unroll
        for (int v = 0; v < 8; ++v) {
            int m = v + (half << 3);
            pre[wave][m * 2 + col] = acc[v];
        }
    }
    __builtin_amdgcn_wave_barrier();   // same-wave LDS is in-order; block reordering

    const float bmu = b_mu[0];
    const float bsg = b_sig[0];
    const float inv_sqrt2 = 0.70710678118654752f;

    for (int r = 0; r < 16; ++r) {
        // All lanes broadcast-read this row's pre-activations.
        float mu_pre  = pre[wave][r * 2 + 0];
        float sig_pre = pre[wave][r * 2 + 1];

        float mu = tanhf(mu_pre + bmu);
        float z  = sig_pre + bsg;
        float sp = (z > 20.0f) ? z : log1pf(expf(z));   // softplus
        float sig = fminf(fmaxf(sp, 0.01f), 100.0f);
        float inv_sig = 1.0f / sig;

        // 101 CDF values, lanes strided by 32.
#pragma unroll
        for (int t = 0; t < 4; ++t) {
            int j = lane + 32 * t;
            if (j <= NCAT) {
                float th = (float)j * 0.02f - 1.0f;     // [-1, 1], C+1 points
                float zz = (th - mu) * inv_sig;
                cdfbuf[wave][j] = 0.5f * erff(zz * inv_sqrt2) + 0.5f;
            }
        }
        __builtin_amdgcn_wave_barrier();

        // sum(probs) telescopes to cdf[C] - cdf[0].
        float invT = 1.0f / (cdfbuf[wave][NCAT] - cdfbuf[wave][0]);

        float* orow = out + (size_t)(row0 + r) * NCAT;
#pragma unroll
        for (int t = 0; t < 4; ++t) {
            int j = lane + 32 * t;
            if (j < NCAT) {
                float p = (cdfbuf[wave][j + 1] - cdfbuf[wave][j]) * invT;
                __builtin_nontemporal_store(p, &orow[j]);
            }
        }
        __builtin_amdgcn_wave_barrier();
    }
}

extern "C" void kernel_launch(void* const* d_in, const int* in_sizes, int n_in,
                              void* d_out, int out_size, void* d_ws, size_t ws_size,
                              hipStream_t stream) {
    const float* x     = (const float*)d_in[0];
    const float* w_mu  = (const float*)d_in[1];
    const float* b_mu  = (const float*)d_in[2];
    const float* w_sig = (const float*)d_in[3];
    const float* b_sig = (const float*)d_in[4];
    float* out = (float*)d_out;

    const int rows = in_sizes[0] / DHID;            // 262144
    const int grid = rows / ROWS_PER_BLOCK;         // 2048 blocks of 256 threads

    unimodal_normal_wmma_kernel<<<grid, BLOCK, 0, stream>>>(
        x, w_mu, b_mu, w_sig, b_sig, out);
}